// KPConv_60756607369859
// MI455X (gfx1250) — compile-verified
//
#include <hip/hip_runtime.h>
#include <math.h>

// KPConv: N=200000 pts, K=9 kernel points, 64->64 ch, M=50000 segments.
// Phase A: guarded atomic scatter of w[k]*x[p,c] into agg[M,576] (almost all
//          weights are zero for N(0,1) points vs 0.15 kernel extent).
// Phase B: out[M,64] = agg[M,576] x W[576,64] with V_WMMA_F32_16X16X4_F32.
//          One wave per 16-row M-tile computing all four 16-col N-tiles
//          (4 accumulators) -> agg is streamed from HBM exactly once.
//          W (144KB) is staged into LDS per block via async global->LDS.

#define NPTS   200000
#define MSEG   50000
#define KPTS   9
#define CIN    64
#define COUT   64
#define KC     (KPTS * CIN)          /* 576 = GEMM reduction dim */
#define MT     (MSEG / 16)           /* 3125 M-tiles */
#define INV_EXT (1.0f / 0.15f)
#define W_BYTES (KC * COUT * 4)      /* 147456 B of LDS for W */

typedef __attribute__((ext_vector_type(2))) float v2f;
typedef __attribute__((ext_vector_type(8))) float v8f;
typedef __attribute__((ext_vector_type(4))) int   v4i;

#if __has_builtin(__builtin_amdgcn_global_load_async_to_lds_b128)
#define HAVE_ASYNC_LDS 1
#endif

static __device__ __forceinline__ void wait_asynccnt0() {
#if __has_builtin(__builtin_amdgcn_s_wait_asynccnt)
    __builtin_amdgcn_s_wait_asynccnt(0);
#else
    asm volatile("s_wait_asynccnt 0x0" ::: "memory");
#endif
}

// ---------------- Phase A: weighted scatter into agg[M, 576] ----------------
__global__ __launch_bounds__(256) void kpconv_scatter(
    const float* __restrict__ s_pts,   // [N,3]
    const float* __restrict__ x,       // [N,64]
    const int*   __restrict__ unq_inv, // [N]
    const float* __restrict__ kp,      // [9,3]
    float*       __restrict__ agg)     // [M,576]
{
    int idx = blockIdx.x * 256 + threadIdx.x;
    if (idx >= NPTS * CIN) return;
    const int p = idx >> 6;
    const int c = idx & 63;

    const float sx = s_pts[p * 3 + 0];
    const float sy = s_pts[p * 3 + 1];
    const float sz = s_pts[p * 3 + 2];

    float w[KPTS];
    bool any = false;
#pragma unroll
    for (int k = 0; k < KPTS; ++k) {
        const float dx = sx - kp[k * 3 + 0];
        const float dy = sy - kp[k * 3 + 1];
        const float dz = sz - kp[k * 3 + 2];
        const float d2 = dx * dx + dy * dy + dz * dz;
        float wk = fmaxf(1.0f - sqrtf(d2) * INV_EXT, 0.0f);
        w[k] = wk;
        any = any || (wk > 0.0f);
    }
    if (!any) return;   // >99.9% of threads exit: no x load, no atomics

    const float xv = x[(size_t)p * CIN + c];
    float* dst = agg + (size_t)unq_inv[p] * KC + c;
#pragma unroll
    for (int k = 0; k < KPTS; ++k) {
        if (w[k] > 0.0f)
            atomicAdd(dst + k * CIN, w[k] * xv);
    }
}

// ---------------- Phase B: out = agg x W via f32 WMMA -----------------------
// 16x16x4 f32 operand layout (ISA 7.12.2):
//   A (16x4): lanes 0-15 -> M=lane, VGPRs = K{0,1}; lanes 16-31 -> K{2,3}
//   B (4x16): lanes 0-15 -> N=lane, rows K{0,1};   lanes 16-31 -> rows K{2,3}
//   C/D     : VGPR r -> M = r + 8*(lane>>4), N = lane&15
__global__ __launch_bounds__(256) void kpconv_gemm(
    const float* __restrict__ agg,  // [M,576]
    const float* __restrict__ W,    // [576,64]
    float*       __restrict__ out)  // [M,64]
{
    extern __shared__ float Wlds[];             // [576][64] = 147456 B
    const int tid = threadIdx.x;

    // ---- Stage W into LDS (async global->LDS when available) ----
#if defined(HAVE_ASYNC_LDS)
    for (int i = tid; i < (KC * COUT) / 4; i += 256) {
        __builtin_amdgcn_global_load_async_to_lds_b128(
            (v4i*)(W + i * 4),
            (__attribute__((address_space(3))) v4i*)(Wlds + i * 4),
            /*offset=*/0, /*cpol=*/0);
    }
    wait_asynccnt0();
#else
    for (int i = tid; i < (KC * COUT) / 4; i += 256) {
        const float4 v = ((const float4*)W)[i];
        ((float4*)Wlds)[i] = v;
    }
#endif
    __syncthreads();

    const int lane = tid & 31;
    const int wave = tid >> 5;
    const int mt   = blockIdx.x * 8 + wave;
    if (mt >= MT) return;                       // wave-uniform exit (after barrier)

    const int half = lane >> 4;                 // 0: K{0,1}, 1: K{2,3}
    const int l15  = lane & 15;
    const int kk   = half * 2;

    const float* arow = agg + (size_t)(mt * 16 + l15) * KC;  // one M row / lane

    v8f c0 = {}, c1 = {}, c2 = {}, c3 = {};
#pragma unroll 2
    for (int k = 0; k < KC; k += 4) {
        v2f a = *(const v2f*)(arow + k + kk);   // 8B-aligned global b64 load
        const float* wr0 = Wlds + (k + kk) * COUT;   // LDS row K=k+kk
        const float* wr1 = wr0 + COUT;               // LDS row K=k+kk+1
        v2f b0, b1, b2, b3;
        b0.x = wr0[l15 +  0]; b0.y = wr1[l15 +  0];
        b1.x = wr0[l15 + 16]; b1.y = wr1[l15 + 16];
        b2.x = wr0[l15 + 32]; b2.y = wr1[l15 + 32];
        b3.x = wr0[l15 + 48]; b3.y = wr1[l15 + 48];
        c0 = __builtin_amdgcn_wmma_f32_16x16x4_f32(false, a, false, b0, (short)0, c0, false, false);
        c1 = __builtin_amdgcn_wmma_f32_16x16x4_f32(false, a, false, b1, (short)0, c1, false, false);
        c2 = __builtin_amdgcn_wmma_f32_16x16x4_f32(false, a, false, b2, (short)0, c2, false, false);
        c3 = __builtin_amdgcn_wmma_f32_16x16x4_f32(false, a, false, b3, (short)0, c3, false, false);
    }

    float* orow = out + (size_t)(mt * 16 + half * 8) * COUT + l15;
#pragma unroll
    for (int r = 0; r < 8; ++r) {
        orow[(size_t)r * COUT +  0] = c0[r];
        orow[(size_t)r * COUT + 16] = c1[r];
        orow[(size_t)r * COUT + 32] = c2[r];
        orow[(size_t)r * COUT + 48] = c3[r];
    }
}

// ---------------------------------------------------------------------------
extern "C" void kernel_launch(void* const* d_in, const int* in_sizes, int n_in,
                              void* d_out, int out_size, void* d_ws, size_t ws_size,
                              hipStream_t stream)
{
    const float* s_pts   = (const float*)d_in[0];
    const float* x       = (const float*)d_in[1];
    const int*   unq_inv = (const int*)  d_in[2];
    const float* weights = (const float*)d_in[3];   // [9,64,64] == [576,64]
    const float* kpoints = (const float*)d_in[4];   // [9,3]
    float* out = (float*)d_out;
    float* agg = (float*)d_ws;                      // [50000, 576] f32

    const size_t agg_bytes = (size_t)MSEG * KC * sizeof(float);  // 115.2 MB
    (void)hipMemsetAsync(agg, 0, agg_bytes, stream);

    const int totalA = NPTS * CIN;
    kpconv_scatter<<<(totalA + 255) / 256, 256, 0, stream>>>(
        s_pts, x, unq_inv, kpoints, agg);

    const int blocks = (MT + 7) / 8;                // 8 M-tiles (waves) / block
    kpconv_gemm<<<blocks, 256, W_BYTES, stream>>>(agg, weights, out);
}